// SparseROIVoxelization_27470610826028
// MI455X (gfx1250) — compile-verified
//
#include <hip/hip_runtime.h>
#include <stdint.h>

// ---------------- problem constants (from reference) ----------------
#define OX 12
#define OY 12
#define OZ 12
#define TOTAL (OX * OY * OZ)   // 1728 voxels per ROI
#define MAXV 128               // sparse output capacity
#define NC 64                  // feature channels
#define NTHREADS 256           // 8 wave32s
#define NWAVES (NTHREADS / 32)
#define CHPTS 256              // points per wave per TDM chunk (768 dwords = 3KB)
#define PLIST_CAP 6144         // in-box point list capacity (expected ~100-300)

#if defined(__has_builtin)
#if __has_builtin(__builtin_amdgcn_tensor_load_to_lds) && \
    __has_builtin(__builtin_amdgcn_s_wait_tensorcnt)
#define USE_TDM 1
#endif
#endif

typedef unsigned int u32;
typedef u32 v4u __attribute__((ext_vector_type(4)));
typedef int v4i __attribute__((ext_vector_type(4)));
typedef int v8i __attribute__((ext_vector_type(8)));

// Order-preserving float <-> uint map so LDS integer atomic max (ds_max_u32)
// implements float max. init 0 == "smaller than any real float".
__device__ __forceinline__ u32 fmap(float f) {
  u32 u = __float_as_uint(f);
  return (u & 0x80000000u) ? ~u : (u | 0x80000000u);
}
__device__ __forceinline__ float funmap(u32 u) {
  return (u & 0x80000000u) ? __uint_as_float(u & 0x7FFFFFFFu)
                           : __uint_as_float(~u);
}

#ifdef USE_TDM
// Issue a 1D TDM copy of `ndw` dwords from global -> LDS byte offset.
// D# packing per CDNA5 ISA:
//  g0: [1:0]=count=1, [63:32]=lds_addr, [120:64]=global_addr, [127:126]=type=2
//  g1: data_size=2 (4B), tensor_dim0=ndw, tensor_dim1=1, tile_dim0=ndw,
//      tile_dim1/2=0, tensor_dim0_stride=ndw
// amdgpu-toolchain (clang-23) builtin arity: 6 args
//  (uint32x4 g0, int32x8 g1, int32x4 g2, int32x4 g3, int32x8 extra, i32 cpol)
__device__ __forceinline__ void tdm_load_1d(const void* gsrc, u32 lds_off,
                                            int ndw) {
  uint64_t ga = (uint64_t)(uintptr_t)gsrc;
  v4u g0;
  g0.x = 1u;                                   // count=1, gather off
  g0.y = lds_off;                              // lds_addr (bytes)
  g0.z = (u32)(ga & 0xFFFFFFFFu);              // global_addr[31:0]
  g0.w = (u32)((ga >> 32) & 0x01FFFFFFu) | 0x80000000u;  // addr[56:32]|type=2
  v8i g1;
  g1[0] = 0x00020000;                          // workgroup_mask=0, data_size=4B
  g1[1] = (int)(((u32)ndw & 0xFFFFu) << 16);   // tensor_dim0[15:0]
  g1[2] = (int)((((u32)ndw >> 16) & 0xFFFFu) | (1u << 16)); // dim0 hi | dim1=1
  g1[3] = (int)(((u32)ndw & 0xFFFFu) << 16);   // tile_dim0 = ndw
  g1[4] = 0;                                   // tile_dim1=0, tile_dim2=0
  g1[5] = ndw;                                 // tensor_dim0_stride[31:0]
  g1[6] = 0;
  g1[7] = 0;
  v4i z4 = {0, 0, 0, 0};
  v8i z8 = {0, 0, 0, 0, 0, 0, 0, 0};
  __builtin_amdgcn_tensor_load_to_lds(g0, g1, z4, z4, z8, 0);
}
#endif

extern "C" __global__ __launch_bounds__(NTHREADS) void roiaware_sparse_pool(
    const float* __restrict__ rois, const float* __restrict__ pts,
    const float* __restrict__ feat, float* __restrict__ out_f,
    int* __restrict__ out_c, int N, int P) {
  __shared__ float ptile[2][NWAVES][CHPTS * 3];  // TDM staging, double buffered
  __shared__ u32 occ[TOTAL];
  __shared__ int slotmap[TOTAL];
  __shared__ u32 sfeat[MAXV * NC];  // mapped bits for float max
  __shared__ u32 scanbuf[NTHREADS];
  __shared__ u32 plist[PLIST_CAP];  // (flat<<16)|pidx of in-box points
  __shared__ u32 plist_cnt, overflow, nocc_sh;

  const int roi = blockIdx.x;
  const int tid = threadIdx.x;
  const int wave = tid >> 5;
  const int lane = tid & 31;
  (void)wave;
  (void)lane;

  // ---- ROI params (scalar, uniform across block) ----
  const float* R = rois + roi * 7;
  const float cx = R[0], cy = R[1], dx = R[3], dy = R[4], dz = R[5];
  const float cz = R[2] + 0.5f * dz;  // bottom-center -> center z
  const float rz = R[6];
  const float cosa = cosf(-rz), sina = sinf(-rz);
  const float hx = 0.5f * dx, hy = 0.5f * dy, hz = 0.5f * dz;
  const float sx = (float)OX / dx, sy = (float)OY / dy, sz = (float)OZ / dz;

  // ---- phase 0: init LDS + outputs ----
  for (int i = tid; i < TOTAL; i += NTHREADS) occ[i] = 0u;
  for (int i = tid; i < MAXV * NC; i += NTHREADS) sfeat[i] = 0u;
  if (tid == 0) {
    plist_cnt = 0u;
    overflow = 0u;
    nocc_sh = 0u;
  }
  float4* of4 = (float4*)(out_f + (size_t)roi * MAXV * NC);
  for (int i = tid; i < MAXV * NC / 4; i += NTHREADS)
    of4[i] = make_float4(0.f, 0.f, 0.f, 0.f);
  int* oc = out_c + (size_t)roi * MAXV * 3;
  for (int i = tid; i < MAXV * 3; i += NTHREADS) oc[i] = -1;
  __syncthreads();

  // point classification, shared by both streaming paths
  auto process = [&](int pidx, float x, float y, float z) {
    float px = x - cx, py = y - cy, pz = z - cz;
    float lx = px * cosa - py * sina;
    float ly = px * sina + py * cosa;
    if ((fabsf(lx) < hx) && (fabsf(ly) < hy) && (fabsf(pz) <= hz)) {
      int vx = (int)floorf((lx + hx) * sx);
      int vy = (int)floorf((ly + hy) * sy);
      int vz = (int)floorf((pz + hz) * sz);
      vx = min(max(vx, 0), OX - 1);
      vy = min(max(vy, 0), OY - 1);
      vz = min(max(vz, 0), OZ - 1);
      int flat = vx * (OY * OZ) + vy * OZ + vz;
      occ[flat] = 1u;
      u32 pos = atomicAdd(&plist_cnt, 1u);
      if (pos < PLIST_CAP)
        plist[pos] = ((u32)flat << 16) | (u32)pidx;
      else
        overflow = 1u;
    }
  };

  // ---- phase 1: stream pts, classify, build compacted in-box list ----
  const int span = NWAVES * CHPTS;  // 2048 points / block iteration
  const int niter = P / span;
#ifdef USE_TDM
  {
    const int base0 = wave * CHPTS;
    const u32 lds0 = (u32)(uintptr_t)&ptile[0][wave][0];
    const u32 lds1 = (u32)(uintptr_t)&ptile[1][wave][0];
    if (niter > 0) tdm_load_1d(pts + (size_t)base0 * 3, lds0, CHPTS * 3);
    for (int it = 0; it < niter; ++it) {
      if (it + 1 < niter) {  // prefetch next chunk into the other buffer
        int nbase = base0 + (it + 1) * span;
        tdm_load_1d(pts + (size_t)nbase * 3, ((it & 1) ? lds0 : lds1),
                    CHPTS * 3);
        __builtin_amdgcn_s_wait_tensorcnt(1);  // chunk `it` has landed
      } else {
        __builtin_amdgcn_s_wait_tensorcnt(0);
      }
      const float* tp = &ptile[it & 1][wave][0];
      const int pbase = base0 + it * span;
      for (int j = lane; j < CHPTS; j += 32)
        process(pbase + j, tp[j * 3 + 0], tp[j * 3 + 1], tp[j * 3 + 2]);
    }
  }
#else
  for (int p = tid; p < niter * span; p += NTHREADS)
    process(p, pts[(size_t)p * 3], pts[(size_t)p * 3 + 1],
            pts[(size_t)p * 3 + 2]);
#endif
  // tail (P not divisible by span): direct global loads
  for (int p = niter * span + tid; p < P; p += NTHREADS)
    process(p, pts[(size_t)p * 3], pts[(size_t)p * 3 + 1],
            pts[(size_t)p * 3 + 2]);
  __syncthreads();

  // ---- phase 2: block scan over 1728 occupancy flags -> slot map + coords ----
  {
    const int IT = 7;  // 256*7 = 1792 >= 1728
    const int base = tid * IT;
    u32 pref[IT];
    u32 s = 0;
    for (int k = 0; k < IT; ++k) {
      int v = base + k;
      pref[k] = s;
      s += (v < TOTAL && occ[v]) ? 1u : 0u;
    }
    scanbuf[tid] = s;
    __syncthreads();
    for (int off = 1; off < NTHREADS; off <<= 1) {  // Hillis-Steele inclusive
      u32 x = scanbuf[tid];
      u32 y = (tid >= off) ? scanbuf[tid - off] : 0u;
      __syncthreads();
      scanbuf[tid] = x + y;
      __syncthreads();
    }
    u32 prev = (tid > 0) ? scanbuf[tid - 1] : 0u;
    if (tid == NTHREADS - 1) nocc_sh = scanbuf[NTHREADS - 1];
    for (int k = 0; k < IT; ++k) {
      int v = base + k;
      if (v < TOTAL) {
        int slot = -1;
        if (occ[v]) {
          u32 r = prev + pref[k];
          if (r < MAXV) slot = (int)r;
        }
        slotmap[v] = slot;
        if (slot >= 0) {
          oc[slot * 3 + 0] = v / (OY * OZ);
          oc[slot * 3 + 1] = (v / OZ) % OY;
          oc[slot * 3 + 2] = v % OZ;
        }
      }
    }
  }
  __syncthreads();

  // ---- phase 3: sparse feature max-pool via LDS ds_max_u32 atomics ----
  if (overflow == 0u) {
    u32 cnt = plist_cnt;
    if (cnt > PLIST_CAP) cnt = PLIST_CAP;
    for (u32 i = tid; i < cnt; i += NTHREADS) {
      u32 e = plist[i];
      int slot = slotmap[e >> 16];
      if (slot >= 0) {
        const float4* f4 = (const float4*)(feat + (size_t)(e & 0xFFFFu) * NC);
        u32* sb = &sfeat[slot * NC];
#pragma unroll
        for (int j = 0; j < NC / 4; ++j) {
          float4 v = f4[j];
          atomicMax(&sb[j * 4 + 0], fmap(v.x));
          atomicMax(&sb[j * 4 + 1], fmap(v.y));
          atomicMax(&sb[j * 4 + 2], fmap(v.z));
          atomicMax(&sb[j * 4 + 3], fmap(v.w));
        }
      }
    }
  } else {  // deterministic fallback: rescan every point (pathological only)
    for (int p = tid; p < P; p += NTHREADS) {
      float x = pts[(size_t)p * 3], y = pts[(size_t)p * 3 + 1],
            z = pts[(size_t)p * 3 + 2];
      float px = x - cx, py = y - cy, pz = z - cz;
      float lx = px * cosa - py * sina;
      float ly = px * sina + py * cosa;
      if ((fabsf(lx) < hx) && (fabsf(ly) < hy) && (fabsf(pz) <= hz)) {
        int vx = min(max((int)floorf((lx + hx) * sx), 0), OX - 1);
        int vy = min(max((int)floorf((ly + hy) * sy), 0), OY - 1);
        int vz = min(max((int)floorf((pz + hz) * sz), 0), OZ - 1);
        int slot = slotmap[vx * (OY * OZ) + vy * OZ + vz];
        if (slot >= 0) {
          const float4* f4 = (const float4*)(feat + (size_t)p * NC);
          u32* sb = &sfeat[slot * NC];
#pragma unroll
          for (int j = 0; j < NC / 4; ++j) {
            float4 v = f4[j];
            atomicMax(&sb[j * 4 + 0], fmap(v.x));
            atomicMax(&sb[j * 4 + 1], fmap(v.y));
            atomicMax(&sb[j * 4 + 2], fmap(v.z));
            atomicMax(&sb[j * 4 + 3], fmap(v.w));
          }
        }
      }
    }
  }
  __syncthreads();

  // ---- phase 4: decode occupied slots to output (rest pre-zeroed) ----
  u32 nocc = nocc_sh;
  if (nocc > MAXV) nocc = MAXV;
  float* ofp = out_f + (size_t)roi * MAXV * NC;
  for (int i = tid; i < (int)(nocc * NC); i += NTHREADS)
    ofp[i] = funmap(sfeat[i]);
}

extern "C" void kernel_launch(void* const* d_in, const int* in_sizes, int n_in,
                              void* d_out, int out_size, void* d_ws,
                              size_t ws_size, hipStream_t stream) {
  (void)n_in;
  (void)out_size;
  (void)d_ws;
  (void)ws_size;
  const float* rois = (const float*)d_in[0];  // [N,7] f32
  const float* pts = (const float*)d_in[1];   // [P,3] f32
  const float* feat = (const float*)d_in[2];  // [P,64] f32
  int N = in_sizes[0] / 7;
  int P = in_sizes[1] / 3;
  float* out_f = (float*)d_out;                          // [N,128,64] f32
  int* out_c = (int*)(out_f + (size_t)N * MAXV * NC);    // [N,128,3] i32
  hipLaunchKernelGGL(roiaware_sparse_pool, dim3(N), dim3(NTHREADS), 0, stream,
                     rois, pts, feat, out_f, out_c, N, P);
}